// HistogramLoss_77773267796135
// MI455X (gfx1250) — compile-verified
//
#include <hip/hip_runtime.h>
#include <hip/hip_bf16.h>

// ---------------- problem constants ----------------
#define NUM_CLASSES 19
#define F 256          // features
#define D 1024         // directions
#define NPIX 8192      // B*H*W = 2*64*64
#define HW 4096        // 64*64
#define COLCAP 8704    // 8192 + 19*16 padding headroom, rounded up

typedef __attribute__((ext_vector_type(2))) float v2f;
typedef __attribute__((ext_vector_type(8))) float v8f;
typedef __attribute__((ext_vector_type(4))) unsigned int u32x4;
typedef __attribute__((ext_vector_type(8))) int i32x8;
typedef __attribute__((ext_vector_type(4))) int i32x4;

// D = A(16x4) * B(4x16) + C, fp32. Layout (per CDNA5 ISA):
//  A: lanes 0-15 M=lane (K=0,1 in the two regs), lanes 16-31 M=lane-16 (K=2,3)
//  B: reg0: lanes0-15 K=0 N=lane / lanes16-31 K=2 ; reg1: K=1 / K=3
//  C/D: reg r: lanes0-15 -> (M=r, N=lane); lanes16-31 -> (M=r+8, N=lane-16)
static __device__ __forceinline__ v8f wmma_f32x4(v2f a, v2f b, v8f c) {
  return __builtin_amdgcn_wmma_f32_16x16x4_f32(
      /*neg_a=*/false, a, /*neg_b=*/false, b,
      /*c_mod=*/(short)0, c, /*reuse_a=*/false, /*reuse_b=*/false);
}

// Tensor Data Mover: contiguous 1D copy of nelem f32 from global -> LDS.
// D# built per CDNA5 ISA:
//  group0: [1:0]count=1 | lds_addr | global_addr(57b) | type=2
//  group1: data_size=4B(code 2), tensor_dim0=tile_dim0=nelem, tile_dim1=1,
//          tensor_dim0_stride=nelem, no padding / multicast / barrier.
// This toolchain's builtin takes 6 args (g0,g1,g2,g3,extra i32x8,cpol).
static __device__ __forceinline__ void tdm_load_1d(unsigned lds_byte_off,
                                                   const void* gptr,
                                                   unsigned nelem) {
  unsigned long long ga = (unsigned long long)gptr;
  u32x4 g0;
  g0[0] = 1u;                                   // count=1 valid user descriptor
  g0[1] = lds_byte_off;                         // lds_addr (bytes)
  g0[2] = (unsigned)(ga & 0xffffffffu);         // global_addr[31:0]
  g0[3] = (unsigned)((ga >> 32) & 0x1ffffffu)   // global_addr[56:32]
          | (2u << 30);                         // type=2 ("image")
  i32x8 g1;
  g1[0] = (int)(2u << 16);                      // data_size: 4 bytes
  g1[1] = (int)((nelem & 0xffffu) << 16);       // tensor_dim0[15:0] @ bits63:48
  g1[2] = (int)(((nelem >> 16) & 0xffffu)       // tensor_dim0[31:16]
                | (1u << 16));                  // tensor_dim1=1 (low bits)
  g1[3] = (int)((nelem & 0xffffu) << 16);       // tile_dim0 @ bits127:112
  g1[4] = 1;                                    // tile_dim1=1, tile_dim2=0
  g1[5] = (int)nelem;                           // tensor_dim0_stride[31:0]
  g1[6] = 0;                                    // stride hi / dim1_stride lo
  g1[7] = 0;
  i32x4 gz4 = {0, 0, 0, 0};                     // groups 2/3 unused (<=2D)
  i32x8 gz8 = {0, 0, 0, 0, 0, 0, 0, 0};
  __builtin_amdgcn_tensor_load_to_lds(g0, g1, gz4, gz4, gz8, 0);
}

// ---------------- K0: zero fill ----------------
__global__ void zero_k(float* p, int n) {
  int i = blockIdx.x * blockDim.x + threadIdx.x;
  if (i < n) p[i] = 0.0f;
}

// ---------------- K1: count + bucket pixels by class ----------------
__global__ void bucket_k(const int* __restrict__ label, int* __restrict__ cnt,
                         int* __restrict__ bucket) {
  int n = blockIdx.x * blockDim.x + threadIdx.x;
  if (n >= NPIX) return;
  int c = label[n];
  if ((unsigned)c < (unsigned)NUM_CLASSES) {
    int pos = atomicAdd(&cnt[c], 1);
    bucket[c * NPIX + pos] = n;
  }
}

// ---------------- K2: tiny exclusive scan of padded counts ----------------
__global__ void scan_k(const int* __restrict__ cnt, int* __restrict__ colOff,
                       int* __restrict__ padcnt) {
  if (threadIdx.x == 0 && blockIdx.x == 0) {
    int run = 0;
    for (int c = 0; c < NUM_CLASSES; ++c) {
      int pc = (cnt[c] + 15) & ~15;
      padcnt[c] = pc;
      colOff[c] = run;
      run += pc;
    }
  }
}

// ---------------- K3: gather-pack X_c column-major (f contiguous per column) ----------------
__global__ void pack_k(const float* __restrict__ feature,
                       const int* __restrict__ bucket,
                       const int* __restrict__ cnt,
                       const int* __restrict__ colOff,
                       const int* __restrict__ padcnt,
                       float* __restrict__ X) {
  int c = blockIdx.x;
  int j0 = blockIdx.y * 16;
  int f = threadIdx.x;
  int nc = cnt[c], pc = padcnt[c];
  long base = colOff[c];
  if (j0 >= pc) return;
  for (int jj = 0; jj < 16; ++jj) {
    int j = j0 + jj;
    float v = 0.0f;
    if (j < nc) {
      int n = bucket[c * NPIX + j];
      int b = n >> 12;
      int hw = n & (HW - 1);
      v = feature[(size_t)b * F * HW + (size_t)f * HW + hw];
    }
    X[(size_t)(base + j) * F + f] = v;
  }
}

// ---------------- K4: class means ----------------
__global__ void mu_k(const float* __restrict__ X, const int* __restrict__ cnt,
                     const int* __restrict__ colOff, float* __restrict__ mu) {
  int c = blockIdx.x, f = threadIdx.x;
  int nc = cnt[c];
  long base = colOff[c];
  float s = 0.0f;
  for (int j = 0; j < nc; ++j) s += X[(size_t)(base + j) * F + f];
  mu[c * F + f] = s / fmaxf((float)nc, 1.0f);
}

// ---------------- K5: covariance tiles via WMMA f32 16x16x4 ----------------
// cov[c] = X_c X_c^T / ns - mu mu^T ; grid (C, 16, 16), one wave per 16x16 tile
__global__ void __launch_bounds__(32) cov_k(const float* __restrict__ X,
                                            const int* __restrict__ cnt,
                                            const int* __restrict__ colOff,
                                            const int* __restrict__ padcnt,
                                            const float* __restrict__ mu,
                                            float* __restrict__ cov) {
  int c = blockIdx.x;
  int f0 = blockIdx.y * 16, g0 = blockIdx.z * 16;
  int lane = threadIdx.x, half = lane >> 4, l = lane & 15;
  int nc = cnt[c], pc = padcnt[c];
  long base = colOff[c];
  v8f acc = {};
  for (int k = 0; k < pc; k += 4) {
    long col = base + k + 2 * half;
    if (k + 8 < pc) __builtin_prefetch(&X[(col + 8) * F + f0 + l], 0, 0);
    v2f a, b;
    a.x = X[col * F + f0 + l];
    a.y = X[(col + 1) * F + f0 + l];
    b.x = X[col * F + g0 + l];
    b.y = X[(col + 1) * F + g0 + l];
    acc = wmma_f32x4(a, b, acc);
  }
  float ns = fmaxf((float)nc, 1.0f);
#pragma unroll
  for (int r = 0; r < 8; ++r) {
    int fr = f0 + half * 8 + r;
    int gc = g0 + l;
    cov[(size_t)c * F * F + (size_t)fr * F + gc] =
        acc[r] / ns - mu[c * F + fr] * mu[c * F + gc];
  }
}

// ---------------- K6: parallel cyclic Jacobi eigensolver, per-class ----------------
// A (256x256 fp32 = 256KB) in dynamic LDS (CDNA5: 320KB/WGP). V in global.
__global__ void __launch_bounds__(256) jacobi_k(const float* __restrict__ cov,
                                                float* __restrict__ Vg,
                                                const float* __restrict__ mu,
                                                float* __restrict__ zmu,
                                                int* __restrict__ order) {
  extern __shared__ float sm[];
  float* A = sm;                       // F*F floats
  float* shC = sm + F * F;             // 128
  float* shS = shC + 128;              // 128
  int* shP = (int*)(shS + 128);        // 128
  int* shQ = shP + 128;                // 128
  float* ev = (float*)(shQ + 128);     // 256
  int* eidx = (int*)(ev + 256);        // 256
  int c = blockIdx.x, t = threadIdx.x;
  const float* covc = cov + (size_t)c * F * F;
  float* Vc = Vg + (size_t)c * F * F;
  for (int i = t; i < F * F; i += 256) {
    A[i] = covc[i];
    Vc[i] = ((i >> 8) == (i & 255)) ? 1.0f : 0.0f;
  }
  __syncthreads();
  for (int sweep = 0; sweep < 8; ++sweep) {
    for (int r = 0; r < 255; ++r) {
      if (t < 128) {
        // round-robin tournament: player 255 fixed
        int p, q;
        if (t == 0) { p = 255; q = r % 255; }
        else { p = (r + t) % 255; q = (r + 255 - t) % 255; }
        if (p > q) { int tmp = p; p = q; q = tmp; }
        float app = A[p * F + p], aqq = A[q * F + q], apq = A[p * F + q];
        float cc = 1.0f, ss = 0.0f;
        if (fabsf(apq) > 1e-20f) {
          float tau = (aqq - app) / (2.0f * apq);
          float tt = copysignf(1.0f, tau) / (fabsf(tau) + sqrtf(1.0f + tau * tau));
          cc = rsqrtf(1.0f + tt * tt);
          ss = tt * cc;
        }
        shP[t] = p; shQ[t] = q; shC[t] = cc; shS[t] = ss;
      }
      __syncthreads();
      // row phase: A <- J^T A (128 disjoint pairs)
      for (int idx = t; idx < 128 * 256; idx += 256) {
        int k = idx >> 8, j = idx & 255;
        int p = shP[k], q = shQ[k];
        float cc = shC[k], ss = shS[k];
        float ap = A[p * F + j], aq = A[q * F + j];
        A[p * F + j] = cc * ap + ss * aq;
        A[q * F + j] = -ss * ap + cc * aq;
      }
      __syncthreads();
      // col phase: A <- A J
      for (int idx = t; idx < 128 * 256; idx += 256) {
        int k = idx >> 8, i2 = idx & 255;
        int p = shP[k], q = shQ[k];
        float cc = shC[k], ss = shS[k];
        float ap = A[i2 * F + p], aq = A[i2 * F + q];
        A[i2 * F + p] = cc * ap + ss * aq;
        A[i2 * F + q] = -ss * ap + cc * aq;
      }
      __syncthreads();
      // eigenvector accumulation: V <- V J (global, L2-resident)
      for (int idx = t; idx < 128 * 256; idx += 256) {
        int k = idx >> 8, i2 = idx & 255;
        int p = shP[k], q = shQ[k];
        float cc = shC[k], ss = shS[k];
        float vp = Vc[i2 * F + p], vq = Vc[i2 * F + q];
        Vc[i2 * F + p] = cc * vp + ss * vq;
        Vc[i2 * F + q] = -ss * vp + cc * vq;
      }
      __syncthreads();
    }
  }
  // sort eigenpairs descending (even-odd transposition), matching eigvecs[:, ::-1]
  ev[t] = A[t * F + t];
  eidx[t] = t;
  __syncthreads();
  for (int ph = 0; ph < 256; ++ph) {
    if (t < 128) {
      int i2 = 2 * t + (ph & 1);
      if (i2 + 1 < 256 && ev[i2] < ev[i2 + 1]) {
        float te = ev[i2]; ev[i2] = ev[i2 + 1]; ev[i2 + 1] = te;
        int ti = eidx[i2]; eidx[i2] = eidx[i2 + 1]; eidx[i2 + 1] = ti;
      }
    }
    __syncthreads();
  }
  order[c * F + t] = eidx[t];
  // zmu[f'] = E_sorted[:,f'] . mu  (so fv = pm @ (z - zmu))
  int col = eidx[t];
  float s = 0.0f;
  for (int g = 0; g < F; ++g) s += Vc[(size_t)g * F + col] * mu[c * F + g];
  zmu[c * F + t] = s;
}

// ---------------- K7: Z = E_sorted^T X_c - zmu, via WMMA; pad cols forced to 0 ----------------
__global__ void __launch_bounds__(32) zproj_k(const float* __restrict__ X,
                                              const float* __restrict__ Vg,
                                              const int* __restrict__ order,
                                              const float* __restrict__ zmu,
                                              const int* __restrict__ cnt,
                                              const int* __restrict__ colOff,
                                              const int* __restrict__ padcnt,
                                              float* __restrict__ Z) {
  int c = blockIdx.x;
  int f0 = blockIdx.y * 16;
  int j0 = blockIdx.z * 16;
  int pc = padcnt[c];
  if (j0 >= pc) return;
  int nc = cnt[c];
  long base = colOff[c];
  int lane = threadIdx.x, half = lane >> 4, l = lane & 15;
  int myord = order[c * F + f0 + l];  // A rows use lanes' M index
  const float* Vc = Vg + (size_t)c * F * F;
  v8f acc = {};
  for (int k = 0; k < F; k += 4) {
    int kk = k + 2 * half;
    v2f a, b;
    a.x = Vc[(size_t)kk * F + myord];
    a.y = Vc[(size_t)(kk + 1) * F + myord];
    b.x = X[(size_t)(base + j0 + l) * F + kk];
    b.y = X[(size_t)(base + j0 + l) * F + kk + 1];
    acc = wmma_f32x4(a, b, acc);
  }
  int j = j0 + l;
#pragma unroll
  for (int r = 0; r < 8; ++r) {
    int fr = f0 + half * 8 + r;
    float zv = (j < nc) ? (acc[r] - zmu[c * F + fr]) : 0.0f;
    Z[(size_t)(base + j) * F + fr] = zv;
  }
}

// ---------------- K8: FV = pm @ Z via WMMA, Z staged through LDS by the TDM ----------------
// Block = 8 waves; each wave owns a 16-row d-tile, all share one 16-column Z
// chunk (16KB) DMA'd into LDS by wave 0 (tensor_load_to_lds), double-buffered.
#define NSEG 8
#define CHUNK_FLOATS (16 * F)  // 4096 floats = 16KB per buffer
__global__ void __launch_bounds__(256) moments_k(const float* __restrict__ pm,
                                                 const float* __restrict__ Z,
                                                 const int* __restrict__ colOff,
                                                 const int* __restrict__ padcnt,
                                                 float* __restrict__ m1,
                                                 float* __restrict__ m2,
                                                 float* __restrict__ m3,
                                                 float* __restrict__ m4) {
  extern __shared__ float lbuf[];  // 2 * CHUNK_FLOATS
  int c = blockIdx.x;
  int wave = threadIdx.x >> 5;
  int lane = threadIdx.x & 31;
  int d0 = (blockIdx.y * 8 + wave) * 16;
  int seg = blockIdx.z;
  int pc = padcnt[c];
  int nchunk = pc >> 4;
  long base = colOff[c];
  int half = lane >> 4, l = lane & 15;
  unsigned ldsbase = (unsigned)__builtin_amdgcn_groupstaticsize();
  v8f s1 = {}, s2 = {}, s3 = {}, s4 = {};
  // prologue: DMA first chunk into buffer 0
  if (seg < nchunk && wave == 0) {
    tdm_load_1d(ldsbase, Z + (size_t)(base + seg * 16) * F, CHUNK_FLOATS);
    __builtin_amdgcn_s_wait_tensorcnt(0);
  }
  __syncthreads();
  int buf = 0;
  for (int ch = seg; ch < nchunk; ch += NSEG) {
    int next = ch + NSEG;
    // kick off DMA of the next chunk into the other buffer
    if (wave == 0 && next < nchunk) {
      tdm_load_1d(ldsbase + (unsigned)((buf ^ 1) * CHUNK_FLOATS * 4),
                  Z + (size_t)(base + next * 16) * F, CHUNK_FLOATS);
    }
    const float* zb = lbuf + buf * CHUNK_FLOATS;  // [col][f], col-major
    v8f acc = {};
    for (int k = 0; k < F; k += 4) {
      int kk = k + 2 * half;
      v2f a = *(const v2f*)&pm[(size_t)(d0 + l) * F + kk];  // global, L1-hot
      v2f b = *(const v2f*)&zb[l * F + kk];                 // LDS (ds_load_b64)
      acc = wmma_f32x4(a, b, acc);
    }
#pragma unroll
    for (int r = 0; r < 8; ++r) {
      float fv = acc[r];
      float fv2 = fv * fv;
      s1[r] += fv;
      s2[r] += fv2;
      s3[r] += fv2 * fv;
      s4[r] += fv2 * fv2;
    }
    if (wave == 0 && next < nchunk) __builtin_amdgcn_s_wait_tensorcnt(0);
    __syncthreads();
    buf ^= 1;
  }
  // reduce the 16 columns held across each 16-lane half (wave32)
#pragma unroll
  for (int r = 0; r < 8; ++r) {
    float a1 = s1[r], a2 = s2[r], a3 = s3[r], a4 = s4[r];
    for (int mmask = 1; mmask < 16; mmask <<= 1) {
      a1 += __shfl_xor(a1, mmask, 16);
      a2 += __shfl_xor(a2, mmask, 16);
      a3 += __shfl_xor(a3, mmask, 16);
      a4 += __shfl_xor(a4, mmask, 16);
    }
    if (l == 0) {
      int d = d0 + half * 8 + r;
      atomicAdd(&m1[c * D + d], a1);
      atomicAdd(&m2[c * D + d], a2);
      atomicAdd(&m3[c * D + d], a3);
      atomicAdd(&m4[c * D + d], a4);
    }
  }
}

// ---------------- K9: kurtosis + per-class loss ----------------
__global__ void __launch_bounds__(256) fin_k(const float* __restrict__ m1,
                                             const float* __restrict__ m2,
                                             const float* __restrict__ m3,
                                             const float* __restrict__ m4,
                                             const int* __restrict__ cnt,
                                             float* __restrict__ numden) {
  __shared__ float red[256];
  int c = blockIdx.x, t = threadIdx.x;
  float ns = fmaxf((float)cnt[c], 1.0f);
  float acc = 0.0f;
  for (int d = t; d < D; d += 256) {
    float a1 = m1[c * D + d] / ns;
    float a2 = m2[c * D + d] / ns;
    float a3 = m3[c * D + d] / ns;
    float a4 = m4[c * D + d] / ns;
    float var = fmaxf(a2 - a1 * a1, 1e-16f);
    float m4c = -3.0f * a1 * a1 * a1 * a1 + 6.0f * a2 * a1 * a1 - 4.0f * a3 * a1 + a4;
    float kurt = m4c / (var * var) - 3.0f;
    kurt = fminf(fmaxf(kurt, -3.0f), 3.0f);
    acc += kurt * kurt;
  }
  red[t] = acc;
  __syncthreads();
  for (int s = 128; s > 0; s >>= 1) {
    if (t < s) red[t] += red[t + s];
    __syncthreads();
  }
  if (t == 0 && cnt[c] > 0) {
    atomicAdd(&numden[0], red[0] / (float)D);
    atomicAdd(&numden[1], 1.0f);
  }
}

__global__ void out_k(const float* __restrict__ numden, float* __restrict__ out) {
  if (threadIdx.x == 0 && blockIdx.x == 0)
    out[0] = numden[0] / fmaxf(numden[1], 1.0f);
}

// ---------------- host launch ----------------
extern "C" void kernel_launch(void* const* d_in, const int* in_sizes, int n_in,
                              void* d_out, int out_size, void* d_ws, size_t ws_size,
                              hipStream_t stream) {
  const float* feature = (const float*)d_in[0];
  const int* label = (const int*)d_in[1];
  const float* pm = (const float*)d_in[2];  // rows already unit-norm (idempotent renorm)
  float* out = (float*)d_out;

  // workspace carve-out (256B aligned sections), ~30 MB total
  char* p = (char*)d_ws;
  auto take = [&](size_t bytes) {
    void* r = (void*)p;
    p += (bytes + 255) & ~(size_t)255;
    return r;
  };
  int* cnt = (int*)take(32 * 4);
  int* colOff = (int*)take(32 * 4);
  int* padcnt = (int*)take(32 * 4);
  float* numden = (float*)take(2 * 4);
  int* bucket = (int*)take((size_t)NUM_CLASSES * NPIX * 4);
  float* mu = (float*)take((size_t)NUM_CLASSES * F * 4);
  float* zmu = (float*)take((size_t)NUM_CLASSES * F * 4);
  int* order = (int*)take((size_t)NUM_CLASSES * F * 4);
  float* X = (float*)take((size_t)COLCAP * F * 4);
  float* Z = (float*)take((size_t)COLCAP * F * 4);
  float* cov = (float*)take((size_t)NUM_CLASSES * F * F * 4);
  float* V = (float*)take((size_t)NUM_CLASSES * F * F * 4);
  float* mom = (float*)take((size_t)4 * NUM_CLASSES * D * 4);  // m1..m4 contiguous
  float* m1 = mom;
  float* m2 = mom + (size_t)NUM_CLASSES * D;
  float* m3 = mom + (size_t)2 * NUM_CLASSES * D;
  float* m4 = mom + (size_t)3 * NUM_CLASSES * D;

  // K0: zero counters, accumulators
  zero_k<<<1, 64, 0, stream>>>((float*)cnt, 32);
  zero_k<<<1, 64, 0, stream>>>(numden, 2);
  zero_k<<<(4 * NUM_CLASSES * D + 255) / 256, 256, 0, stream>>>(mom, 4 * NUM_CLASSES * D);

  // K1: bucket pixels by class
  bucket_k<<<NPIX / 256, 256, 0, stream>>>(label, cnt, bucket);

  // K2: offsets
  scan_k<<<1, 32, 0, stream>>>(cnt, colOff, padcnt);

  // K3: pack X (column-major, pad cols zeroed)
  pack_k<<<dim3(NUM_CLASSES, NPIX / 16, 1), 256, 0, stream>>>(feature, bucket, cnt,
                                                              colOff, padcnt, X);

  // K4: class means
  mu_k<<<NUM_CLASSES, 256, 0, stream>>>(X, cnt, colOff, mu);

  // K5: covariance via WMMA (one wave per 16x16 tile)
  cov_k<<<dim3(NUM_CLASSES, 16, 16), 32, 0, stream>>>(X, cnt, colOff, padcnt, mu, cov);

  // K6: Jacobi eigensolve, 256KB A in LDS + small shared arrays
  size_t lds = (size_t)F * F * 4 + 512 * 4 + 512 * 4;  // 266240 B (<320KB/WGP)
  jacobi_k<<<NUM_CLASSES, 256, lds, stream>>>(cov, V, mu, zmu, order);

  // K7: Z = E^T X - zmu
  zproj_k<<<dim3(NUM_CLASSES, 16, NPIX / 16), 32, 0, stream>>>(X, V, order, zmu, cnt,
                                                               colOff, padcnt, Z);

  // K8: FV tiles + moments, TDM double-buffered Z staging (2x16KB LDS)
  moments_k<<<dim3(NUM_CLASSES, D / (16 * 8), NSEG), 256,
              2 * CHUNK_FLOATS * 4, stream>>>(pm, Z, colOff, padcnt, m1, m2, m3, m4);

  // K9/K10: kurtosis, masked mean, scalar loss
  fin_k<<<NUM_CLASSES, 256, 0, stream>>>(m1, m2, m3, m4, cnt, numden);
  out_k<<<1, 32, 0, stream>>>(numden, out);
}